// GA_MGMC_8211977470522
// MI455X (gfx1250) — compile-verified
//
#include <hip/hip_runtime.h>

typedef float v2f __attribute__((ext_vector_type(2)));
typedef float v8f __attribute__((ext_vector_type(8)));

#define G_GRAPHS 16
#define NPER     128
#define NUNIV    128
#define NTOT     (G_GRAPHS * NPER)   // 2048
#define UELEMS   (NTOT * NUNIV)      // 262144
#define MTILE    64                  // rows of C per wave (4 x 16-row WMMA frags)

// ---------------------------------------------------------------------------
// fp32 WMMA GEMM:  C = alpha * (A @ B) (+ beta * Add)
// One wave (32 threads) computes a 64x16 tile of C: four v8f accumulators
// share the B fragment each K-step, so VMEM/WMMA is 4x8B (A) + 8B (B) per
// 4 WMMAs. Batched via blockIdx.z with per-operand strides (a diagonal
// stride of 128*ld+128 turns this into the block-diagonal GEMM).
//
// Fragment layouts (CDNA5 ISA 7.12.2, wave32):
//   A 16x4 : lanes 0-15 -> M=0..15; VGPR0/1 = K={2*hi, 2*hi+1}
//   B 4x16 : VGPR v, lanes 0-15 -> K=2*hi+v row, N=lane
//   C 16x16: VGPR v, lane -> M = 8*hi + v, N = lane&15
// ---------------------------------------------------------------------------
__global__ void wmma_gemm_f32(float* __restrict__ C, int ldc, long strideC,
                              const float* __restrict__ A, int lda, long strideA,
                              const float* __restrict__ B, int ldb, long strideB,
                              const float* __restrict__ Add, long strideAdd,
                              int K, float alpha, float beta)
{
    const int lane = threadIdx.x;           // 0..31, one wave per block
    const int lo = lane & 15;
    const int hi = lane >> 4;
    const int n0 = blockIdx.x * 16;
    const int m0 = blockIdx.y * MTILE;
    const long z = blockIdx.z;

    A += z * strideA;
    B += z * strideB;
    C += z * strideC;

    const float* Ap = A + (long)(m0 + lo) * lda + 2 * hi;   // row m, cols k0+2hi..
    const float* Bp = B + (long)(2 * hi) * ldb + n0 + lo;   // row k0+2hi, col n
    const long aRow16 = 16L * lda;

    v8f acc0 = {}, acc1 = {}, acc2 = {}, acc3 = {};
    for (int k0 = 0; k0 < K; k0 += 4) {
        if ((k0 & 127) == 0)                 // gfx1250 global_prefetch_b8 on the
            __builtin_prefetch(Ap + k0 + 256, 0, 0);   // W/A stream (L2-resident)
        v2f b;
        b.x = Bp[(long)k0 * ldb];            // B[k0+2hi  ][n]
        b.y = Bp[(long)k0 * ldb + ldb];      // B[k0+2hi+1][n]
        v2f a0 = *(const v2f*)(Ap + k0);
        v2f a1 = *(const v2f*)(Ap + aRow16 + k0);
        v2f a2 = *(const v2f*)(Ap + 2 * aRow16 + k0);
        v2f a3 = *(const v2f*)(Ap + 3 * aRow16 + k0);
        acc0 = __builtin_amdgcn_wmma_f32_16x16x4_f32(false, a0, false, b, (short)0, acc0, false, false);
        acc1 = __builtin_amdgcn_wmma_f32_16x16x4_f32(false, a1, false, b, (short)0, acc1, false, false);
        acc2 = __builtin_amdgcn_wmma_f32_16x16x4_f32(false, a2, false, b, (short)0, acc2, false, false);
        acc3 = __builtin_amdgcn_wmma_f32_16x16x4_f32(false, a3, false, b, (short)0, acc3, false, false);
    }

    const float* Addp = Add ? (Add + z * strideAdd) : nullptr;
    v8f accs[4] = {acc0, acc1, acc2, acc3};
#pragma unroll
    for (int r = 0; r < 4; ++r) {
#pragma unroll
        for (int v = 0; v < 8; ++v) {
            const int row = m0 + 16 * r + hi * 8 + v;
            const long idx = (long)row * ldc + n0 + lo;
            float val = alpha * accs[r][v];
            if (Addp) val += beta * Addp[idx];
            C[idx] = val;
        }
    }
}

// ---------------------------------------------------------------------------
// C[MxN] = A^T @ B where A is K x M (row-major) — used for T = U^T @ AU.
// Only 128x128 output with K=2048; negligible cost, 16x16 tile per wave.
// ---------------------------------------------------------------------------
__global__ void wmma_gemm_AtB_f32(float* __restrict__ C, int ldc,
                                  const float* __restrict__ A, int lda,
                                  const float* __restrict__ B, int ldb, int K)
{
    const int lane = threadIdx.x;
    const int lo = lane & 15;
    const int hi = lane >> 4;
    const int n0 = blockIdx.x * 16;
    const int m0 = blockIdx.y * 16;

    v8f acc = {};
    for (int k0 = 0; k0 < K; k0 += 4) {
        v2f a;
        a.x = A[(long)(k0 + 2 * hi)     * lda + m0 + lo];   // A^T[m][k] = A[k][m]
        a.y = A[(long)(k0 + 2 * hi + 1) * lda + m0 + lo];
        v2f b;
        b.x = B[(long)(k0 + 2 * hi)     * ldb + n0 + lo];
        b.y = B[(long)(k0 + 2 * hi + 1) * ldb + n0 + lo];
        acc = __builtin_amdgcn_wmma_f32_16x16x4_f32(
            false, a, false, b, (short)0, acc, false, false);
    }
#pragma unroll
    for (int v = 0; v < 8; ++v) {
        const int r = m0 + hi * 8 + v;
        C[(long)r * ldc + n0 + lo] = acc[v];
    }
}

// ---------------------------------------------------------------------------
// Log-space Sinkhorn, one graph per workgroup: the 128x128 slab lives in LDS
// (64 KB of the WGP's 320 KB). 10 iterations alternating row (axis=2) and
// column (axis=1) logsumexp normalization, then U = exp(log_s).
// ---------------------------------------------------------------------------
__global__ void sinkhorn_kernel(const float* __restrict__ V,
                                float* __restrict__ U, float invTau)
{
    __shared__ float x[NPER * NUNIV];       // 64 KB
    const int g = blockIdx.x;
    const int tid = threadIdx.x;            // 256 threads = 8 waves
    const long base = (long)g * NPER * NUNIV;

    for (int idx = tid; idx < NPER * NUNIV; idx += 256)
        x[idx] = V[base + idx] * invTau;
    __syncthreads();

    for (int it = 0; it < 10; ++it) {
        if (tid < NPER) {
            if ((it & 1) == 0) {            // even: normalize over n_univ (rows)
                const int r = tid;
                float m = -3.0e38f;
                for (int j = 0; j < NUNIV; ++j) m = fmaxf(m, x[r * NUNIV + j]);
                float s = 0.0f;
                for (int j = 0; j < NUNIV; ++j) s += expf(x[r * NUNIV + j] - m);
                const float lse = m + logf(s);
                for (int j = 0; j < NUNIV; ++j) x[r * NUNIV + j] -= lse;
            } else {                        // odd: normalize over n (columns)
                const int c = tid;
                float m = -3.0e38f;
                for (int i = 0; i < NPER; ++i) m = fmaxf(m, x[i * NUNIV + c]);
                float s = 0.0f;
                for (int i = 0; i < NPER; ++i) s += expf(x[i * NUNIV + c] - m);
                const float lse = m + logf(s);
                for (int i = 0; i < NPER; ++i) x[i * NUNIV + c] -= lse;
            }
        }
        __syncthreads();
    }

    for (int idx = tid; idx < NPER * NUNIV; idx += 256)
        U[base + idx] = expf(x[idx]);
}

// Second tuple output: zeros(16, int32) — bit pattern identical to 0.0f.
__global__ void tail_zero_kernel(float* __restrict__ out)
{
    if (threadIdx.x < G_GRAPHS) out[UELEMS + threadIdx.x] = 0.0f;
}

extern "C" void kernel_launch(void* const* d_in, const int* in_sizes, int n_in,
                              void* d_out, int out_size, void* d_ws, size_t ws_size,
                              hipStream_t stream)
{
    (void)in_sizes; (void)n_in; (void)out_size; (void)ws_size;
    const float* A  = (const float*)d_in[0];   // [2048,2048] block-diag
    const float* W  = (const float*)d_in[1];   // [2048,2048]
    const float* U0 = (const float*)d_in[2];   // [2048,128]

    float* ws = (float*)d_ws;
    float* U  = ws;                 // [2048,128]
    float* AU = ws + 1L * UELEMS;   // [2048,128]
    float* P  = ws + 2L * UELEMS;   // [2048,128]
    float* WU = ws + 3L * UELEMS;   // [2048,128]
    float* V  = ws + 4L * UELEMS;   // [2048,128]
    float* T  = ws + 5L * UELEMS;   // [128,128]

    hipMemcpyAsync(U, U0, (size_t)UELEMS * sizeof(float),
                   hipMemcpyDeviceToDevice, stream);

    const long diagStride = (long)NPER * NTOT + NPER;   // next 128x128 diag block
    const long slabStride = (long)NPER * NUNIV;         // next graph's 128x128 slab
    const float taus[3] = {0.5f, 0.25f, 0.125f};

    for (int t = 0; t < 3; ++t) {
        for (int it = 0; it < 3; ++it) {
            // AU_g = A_g @ U_g   (16 batched 128x128 GEMMs)
            wmma_gemm_f32<<<dim3(NUNIV/16, NPER/MTILE, G_GRAPHS), 32, 0, stream>>>(
                AU, NUNIV, slabStride,
                A,  NTOT,  diagStride,
                U,  NUNIV, slabStride,
                nullptr, 0, NPER, 1.0f, 0.0f);

            // WU = W @ U   (the hot GEMM: 2048x2048x128, W is L2-resident)
            wmma_gemm_f32<<<dim3(NUNIV/16, NTOT/MTILE, 1), 32, 0, stream>>>(
                WU, NUNIV, 0, W, NTOT, 0, U, NUNIV, 0,
                nullptr, 0, NTOT, 1.0f, 0.0f);

            // T = U^T @ AU   (128x128, K=2048)
            wmma_gemm_AtB_f32<<<dim3(NUNIV/16, NUNIV/16, 1), 32, 0, stream>>>(
                T, NUNIV, U, NUNIV, AU, NUNIV, NTOT);

            // P = U @ T   ( (U U^T) AU  via associativity )
            wmma_gemm_f32<<<dim3(NUNIV/16, NTOT/MTILE, 1), 32, 0, stream>>>(
                P, NUNIV, 0, U, NUNIV, 0, T, NUNIV, 0,
                nullptr, 0, NUNIV, 1.0f, 0.0f);

            // V = (2/G) * blockdiag(A) @ P + (1/G) * WU
            wmma_gemm_f32<<<dim3(NUNIV/16, NPER/MTILE, G_GRAPHS), 32, 0, stream>>>(
                V,  NUNIV, slabStride,
                A,  NTOT,  diagStride,
                P,  NUNIV, slabStride,
                WU, slabStride, NPER,
                2.0f / G_GRAPHS, 1.0f / G_GRAPHS);

            // U = sinkhorn_log(V / tau), 10 iters, per-graph in LDS
            sinkhorn_kernel<<<G_GRAPHS, 256, 0, stream>>>(V, U, 1.0f / taus[t]);
        }
    }

    hipMemcpyAsync(d_out, U, (size_t)UELEMS * sizeof(float),
                   hipMemcpyDeviceToDevice, stream);
    tail_zero_kernel<<<1, 32, 0, stream>>>((float*)d_out);
}